// GAE_Decoder_72035191489094
// MI455X (gfx1250) — compile-verified
//
#include <hip/hip_runtime.h>
#include <hip/hip_bf16.h>

// ---------------------------------------------------------------------------
// GAE decoder + 2-layer GCN for MI455X (gfx1250, wave32, WMMA).
//
//   ew   = sigmoid(<x[row], x[col]>)                       (bandwidth: gathers)
//   deg  = segsum(ew by col) + 1 ; dinv = rsqrt(deg)
//   xw   = x @ W1            (WMMA f32 16x16x4, 16x64 tile/wave)
//   h    = relu(segsum(norm * xw[row] by col) + dinv^2*xw + b1)
//   hw   = h @ W2            (WMMA f32 16x16x4, 16x64 tile/wave)
//   out  = segsum(norm * hw[row] by col) + dinv^2*hw + b2
// ---------------------------------------------------------------------------

#define GN_NODES 50000
#define GN_EDGES 600000
#define GD_IN    128
#define GD_HID   256
#define GD_OUT   128

typedef float  v2f __attribute__((ext_vector_type(2)));
typedef float  v8f __attribute__((ext_vector_type(8)));

// ---------------------------------------------------------------------------
// 1) Edge decode: one wave32 per edge; 32 lanes x float4 == 128 floats/row.
// ---------------------------------------------------------------------------
__global__ __launch_bounds__(256)
void edge_decode_kernel(const float* __restrict__ x,
                        const long long* __restrict__ ei,
                        float* __restrict__ ew, int nE) {
    int e = blockIdx.x * (blockDim.x >> 5) + (threadIdx.x >> 5);
    if (e >= nE) return;
    int lane = threadIdx.x & 31;
    long long r = ei[e];
    long long c = ei[(long long)nE + e];
    const float4* xr = (const float4*)(x + r * (long long)GD_IN);
    const float4* xc = (const float4*)(x + c * (long long)GD_IN);
    float4 a = xr[lane];
    float4 b = xc[lane];
    float s = a.x * b.x + a.y * b.y + a.z * b.z + a.w * b.w;
    #pragma unroll
    for (int off = 16; off > 0; off >>= 1)
        s += __shfl_xor(s, off, 32);
    if (lane == 0)
        ew[e] = 1.0f / (1.0f + __expf(-s));
}

// ---------------------------------------------------------------------------
// 2) Degree: init to 1 (self loop), atomic segment-sum, then rsqrt in place.
// ---------------------------------------------------------------------------
__global__ void deg_init_kernel(float* __restrict__ deg, int n) {
    int i = blockIdx.x * blockDim.x + threadIdx.x;
    if (i < n) deg[i] = 1.0f;
}

__global__ void deg_accum_kernel(const long long* __restrict__ ei,
                                 const float* __restrict__ ew,
                                 float* __restrict__ deg, int nE) {
    int e = blockIdx.x * blockDim.x + threadIdx.x;
    if (e >= nE) return;
    long long c = ei[(long long)nE + e];   // col
    atomicAdd(deg + c, ew[e]);
}

__global__ void dinv_kernel(float* __restrict__ deg, int n) {
    int i = blockIdx.x * blockDim.x + threadIdx.x;
    if (i >= n) return;
    float d = deg[i];
    deg[i] = (d > 0.0f) ? __frsqrt_rn(d) : 0.0f;
}

// ---------------------------------------------------------------------------
// 3) GEMM via V_WMMA_F32_16X16X4_F32, register-blocked: each wave computes a
//    16 x (16*NB) output tile. A fragment loaded once per k-step, reused for
//    NB WMMAs (cuts A traffic NB x; load:wmma ratio 2.25:1 at NB=4).
//    A 16x4 frag: lane m = lane&15, VGPR0/1 hold K = kb, kb+1 (kb = 2*(lane>>4))
//    B 4x16 frag: lane n = lane&15, VGPR0/1 hold rows K = kb, kb+1
//    C/D 16x16 : VGPR j -> row j + 8*(lane>>4), col lane&15
// ---------------------------------------------------------------------------
template<int K, int NCOLS, int NB>
__global__ __launch_bounds__(256)
void gemm_wmma_f32_kernel(const float* __restrict__ X,
                          const float* __restrict__ W,
                          float* __restrict__ Y, int nrows) {
    const int tilesN = NCOLS / (16 * NB);
    int wave = threadIdx.x >> 5;
    int lane = threadIdx.x & 31;
    int tile = blockIdx.x * 8 + wave;
    int tM = tile / tilesN;
    int tN = tile - tM * tilesN;
    if (tM * 16 >= nrows) return;          // wave-uniform; EXEC stays all-ones

    int m  = tM * 16 + (lane & 15);
    int n0 = tN * (16 * NB) + (lane & 15);
    int kb = (lane >> 4) << 1;

    const float* xrow = X + (size_t)m * K + kb;
    v8f acc[NB];
    #pragma unroll
    for (int j = 0; j < NB; ++j) acc[j] = (v8f){};

    #pragma unroll 4
    for (int k = 0; k < K; k += 4) {
        v2f a = *(const v2f*)(xrow + k);                 // A[m][k+kb .. +1]
        const float* wr0 = W + (size_t)(k + kb) * NCOLS + n0;
        const float* wr1 = wr0 + NCOLS;
        #pragma unroll
        for (int j = 0; j < NB; ++j) {
            v2f b;
            b[0] = wr0[j * 16];                          // B[k+kb][n0+16j]
            b[1] = wr1[j * 16];                          // B[k+kb+1][n0+16j]
            acc[j] = __builtin_amdgcn_wmma_f32_16x16x4_f32(
                /*neg_a=*/false, a, /*neg_b=*/false, b,
                /*c_mod=*/(short)0, acc[j], /*reuse_a=*/false, /*reuse_b=*/false);
        }
    }

    int row0 = tM * 16 + ((lane >> 4) << 3);
    int col0 = tN * (16 * NB) + (lane & 15);
    #pragma unroll
    for (int j = 0; j < NB; ++j) {
        #pragma unroll
        for (int r = 0; r < 8; ++r)
            Y[(size_t)(row0 + r) * NCOLS + col0 + j * 16] = acc[j][r];
    }
}

// ---------------------------------------------------------------------------
// 4) Scatter-aggregate: one wave per edge; lane-strided f32 atomics into AGG.
// ---------------------------------------------------------------------------
template<int D>
__global__ __launch_bounds__(256)
void scatter_agg_kernel(const long long* __restrict__ ei,
                        const float* __restrict__ ew,
                        const float* __restrict__ dinv,
                        const float* __restrict__ XW,
                        float* __restrict__ AGG, int nE) {
    int e = blockIdx.x * (blockDim.x >> 5) + (threadIdx.x >> 5);
    if (e >= nE) return;
    int lane = threadIdx.x & 31;
    long long r = ei[e];
    long long c = ei[(long long)nE + e];
    float norm = dinv[r] * ew[e] * dinv[c];
    const float* src = XW + r * (long long)D;
    float* dst = AGG + c * (long long)D;
    #pragma unroll
    for (int d = lane; d < D; d += 32)
        atomicAdd(dst + d, norm * src[d]);
}

// ---------------------------------------------------------------------------
// 5) Finalize: AGG += dinv^2 * XW + bias ; optional ReLU. In place.
// ---------------------------------------------------------------------------
template<int D, bool RELU>
__global__ void finalize_kernel(float* __restrict__ AGG,
                                const float* __restrict__ XW,
                                const float* __restrict__ dinv,
                                const float* __restrict__ bias, int n) {
    long long idx = (long long)blockIdx.x * blockDim.x + threadIdx.x;
    long long total = (long long)n * D;
    if (idx >= total) return;
    int i = (int)(idx / D);
    int d = (int)(idx - (long long)i * D);
    float di = dinv[i];
    float v = AGG[idx] + di * di * XW[idx] + bias[d];
    if (RELU) v = fmaxf(v, 0.0f);
    AGG[idx] = v;
}

// ---------------------------------------------------------------------------
extern "C" void kernel_launch(void* const* d_in, const int* in_sizes, int n_in,
                              void* d_out, int out_size, void* d_ws, size_t ws_size,
                              hipStream_t stream) {
    const float*     x   = (const float*)d_in[0];
    const long long* ei  = (const long long*)d_in[1];   // int64 (2, E)
    const float*     W1  = (const float*)d_in[2];
    const float*     b1  = (const float*)d_in[3];
    const float*     W2  = (const float*)d_in[4];
    const float*     b2  = (const float*)d_in[5];

    float* out_f = (float*)d_out;

    // Output tuple layout: [ out (N*D_OUT) | edge_index (raw copy) | edge_weight (E) ]
    float* ew     = out_f + ((size_t)out_size - GN_EDGES);
    float* ei_out = out_f + (size_t)GN_NODES * GD_OUT;
    size_t ei_bytes =
        ((size_t)out_size - (size_t)GN_NODES * GD_OUT - (size_t)GN_EDGES) * sizeof(float);

    // Workspace: [ dinv 50048 | bufA 50000*256 (xw1, later hw2) | bufB 50000*256 (agg1/h) ]
    float* dinv = (float*)d_ws;
    float* bufA = dinv + 50048;
    float* bufB = bufA + (size_t)GN_NODES * GD_HID;

    const int WPB = 8;  // waves per 256-thread block
    const int NB  = 4;  // 16-col tiles per wave in the GEMM

    // 1) edge weights -> directly into their output slot
    edge_decode_kernel<<<(GN_EDGES + WPB - 1) / WPB, 256, 0, stream>>>(x, ei, ew, GN_EDGES);

    // edge_index passthrough (raw device-to-device copy, graph-capturable)
    if (ei_bytes > 0)
        hipMemcpyAsync(ei_out, (const void*)ei, ei_bytes, hipMemcpyDeviceToDevice, stream);

    // 2) degrees -> dinv
    deg_init_kernel<<<(GN_NODES + 255) / 256, 256, 0, stream>>>(dinv, GN_NODES);
    deg_accum_kernel<<<(GN_EDGES + 255) / 256, 256, 0, stream>>>(ei, ew, dinv, GN_EDGES);
    dinv_kernel<<<(GN_NODES + 255) / 256, 256, 0, stream>>>(dinv, GN_NODES);

    // 3) xw1 = x @ W1   (3125 M-tiles x 4 N-groups = 12500 wave-tiles)
    {
        int tiles  = (GN_NODES / 16) * (GD_HID / (16 * NB));
        int blocks = (tiles + WPB - 1) / WPB;
        gemm_wmma_f32_kernel<GD_IN, GD_HID, NB>
            <<<blocks, 256, 0, stream>>>(x, W1, bufA, GN_NODES);
    }

    // 4) layer-1 aggregate + finalize (ReLU) -> h in bufB
    hipMemsetAsync(bufB, 0, (size_t)GN_NODES * GD_HID * sizeof(float), stream);
    scatter_agg_kernel<GD_HID>
        <<<(GN_EDGES + WPB - 1) / WPB, 256, 0, stream>>>(ei, ew, dinv, bufA, bufB, GN_EDGES);
    {
        long long total = (long long)GN_NODES * GD_HID;
        finalize_kernel<GD_HID, true>
            <<<(unsigned)((total + 255) / 256), 256, 0, stream>>>(bufB, bufA, dinv, b1, GN_NODES);
    }

    // 5) hw2 = h @ W2 -> reuse bufA (only first N*D_OUT floats)
    {
        int tiles  = (GN_NODES / 16) * (GD_OUT / (16 * NB));
        int blocks = (tiles + WPB - 1) / WPB;
        gemm_wmma_f32_kernel<GD_HID, GD_OUT, NB>
            <<<blocks, 256, 0, stream>>>(bufB, W2, bufA, GN_NODES);
    }

    // 6) layer-2 aggregate directly into d_out, finalize in place (no ReLU)
    hipMemsetAsync(out_f, 0, (size_t)GN_NODES * GD_OUT * sizeof(float), stream);
    scatter_agg_kernel<GD_OUT>
        <<<(GN_EDGES + WPB - 1) / WPB, 256, 0, stream>>>(ei, ew, dinv, bufA, out_f, GN_EDGES);
    {
        long long total = (long long)GN_NODES * GD_OUT;
        finalize_kernel<GD_OUT, false>
            <<<(unsigned)((total + 255) / 256), 256, 0, stream>>>(out_f, bufA, dinv, b2, GN_NODES);
    }
}